// SingleNonLinear2d_56401510531308
// MI455X (gfx1250) — compile-verified
//
#include <hip/hip_runtime.h>

typedef __attribute__((ext_vector_type(2))) float v2f;
typedef __attribute__((ext_vector_type(8))) float v8f;

#define DIMSZ 256
#define NBATCH 16
#define NPIX (DIMSZ * DIMSZ)          // 65536
#define NELEM (NBATCH * NPIX)         // 1048576
#define LDSW 138                      // 136 used cols + pad (bank-conflict avoidance)
#define LDSH 22                       // 16 + 6 halo rows
#define NSCHUNK (7 * 6)               // 7 ky rows x 6 K-chunks of 4
#define SBUF_ELEMS (NSCHUNK * 32 * 2) // lane-formatted B operands: 2688 floats

// ---------------------------------------------------------------------------
// Init: fold 15-kernel conv + channel combo into one effective 7x7 kernel,
// then pre-lay-out the banded S matrices in WMMA B-operand lane format:
//   S[k][n] = w_eff[ky][k-n] for 0<=k-n<=6, k<22, else 0   (K padded 22->24)
//   B operand striping (32-bit, 4x16): VGPR v, lane l -> K = 4c + v + 2*(l>>4), N = l&15
// ---------------------------------------------------------------------------
__global__ void fdtd_init(const float* __restrict__ fdk,
                          const float* __restrict__ cw,
                          float* __restrict__ Sbuf) {
    __shared__ float weff[49];
    int t = threadIdx.x;
    if (t < 49) {
        float acc = 0.0f;
        #pragma unroll
        for (int ch = 1; ch < 15; ++ch)
            acc += cw[ch - 1] * fdk[ch * 49 + t];
        // rhs uses (2*dt/mu)*combo with dt=0.5, mu=1 -> factor 1.0 (kept explicit)
        weff[t] = acc * (2.0f * 0.5f / 1.0f);
    }
    __syncthreads();
    for (int e = t; e < SBUF_ELEMS; e += blockDim.x) {
        int v    = e & 1;
        int lane = (e >> 1) & 31;
        int c    = (e >> 6) % 6;
        int ky   = e / 384;
        int n    = lane & 15;
        int k    = 4 * c + v + 2 * (lane >> 4);
        int j    = k - n;
        float val = (k < 22 && j >= 0 && j <= 6) ? weff[ky * 7 + j] : 0.0f;
        Sbuf[e] = val;
    }
}

// ---------------------------------------------------------------------------
// One FDTD time step. Block = 256 threads (8 waves); each wave computes one
// 16x16 output tile of a 128x16 strip via 42x v_wmma_f32_16x16x4_f32.
// Grid: (batch=16, tileY=16, stripX=2).
// ---------------------------------------------------------------------------
__global__ __launch_bounds__(256) void fdtd_step(
    const float* __restrict__ cur,   // u1 at this step
    const float* prev,               // u0 at this step (read at own pixel only)
    const float* __restrict__ epsr,
    const float* __restrict__ sigma,
    const float* __restrict__ Sglob, // lane-formatted banded kernel matrices
    const float* __restrict__ src,   // (32,2)
    float* next,                     // u2 output
    int step) {
    __shared__ float tile[LDSH * LDSW];
    __shared__ float Slds[SBUF_ELEMS];

    const int b  = blockIdx.x;
    const int y0 = blockIdx.y * 16;
    const int x0 = blockIdx.z * 128;
    const int t  = threadIdx.x;
    const float* curb = cur + b * NPIX;

    // Cooperative haloed load of the 128x16 strip (+3 halo, zero padding).
    for (int e = t; e < LDSH * LDSW; e += 256) {
        int r = e / LDSW, c = e % LDSW;
        int gy = y0 + r - 3, gx = x0 + c - 3;
        float v = 0.0f;
        if (c < 136 && gy >= 0 && gy < DIMSZ && gx >= 0 && gx < DIMSZ)
            v = curb[gy * DIMSZ + gx];
        tile[e] = v;
    }
    for (int e = t; e < SBUF_ELEMS; e += 256) Slds[e] = Sglob[e];
    __syncthreads();

    const int wave = t >> 5;
    const int lane = t & 31;
    const int mlo  = lane & 15;  // A/B: M or N = lane&15
    const int sel  = lane >> 4;  // K half selector for A/B striping
    const int xw   = 16 * wave;  // this wave's tile column base within strip

    v8f acc = {0.f, 0.f, 0.f, 0.f, 0.f, 0.f, 0.f, 0.f};
    #pragma unroll
    for (int ky = 0; ky < 7; ++ky) {
        const int arow = (mlo + ky) * LDSW + xw + 2 * sel;
        #pragma unroll
        for (int c = 0; c < 6; ++c) {
            // A operand: lane holds A[mlo][4c + 2*sel + {0,1}] (8B-aligned LDS load)
            v2f a = *(const v2f*)&tile[arow + 4 * c];
            // B operand: precomputed in exact lane format
            v2f bmat = *(const v2f*)&Slds[((ky * 6 + c) * 32 + lane) * 2];
            acc = __builtin_amdgcn_wmma_f32_16x16x4_f32(
                false, a, false, bmat, (short)0, acc, false, false);
        }
    }

    // Pointwise FDTD update fused as epilogue.
    const float dt = 0.5f;
    const float s0 = src[2 * step], s1 = src[2 * step + 1];
    const float srcterm = -(s0 - s1);
    const int gx = x0 + xw + mlo;
    #pragma unroll
    for (int r = 0; r < 8; ++r) {
        int m  = r + 8 * sel;          // C/D striping: VGPR r -> M = r + 8*(lane>=16)
        int gy = y0 + m;
        int idx = b * NPIX + gy * DIMSZ + gx;
        float ep = epsr[idx], sg = sigma[idx];
        float inv_den = 1.0f / (2.0f * ep / dt + sg);
        float av = 4.0f * ep / dt;
        float bv = 2.0f * ep / dt - sg;
        float cu = tile[(m + 3) * LDSW + xw + mlo + 3]; // center u1 from LDS
        float pv = prev[idx];
        float rhs = acc[r] + av * cu - bv * pv;         // 2dt/mu folded into w_eff
        if (gy == 128 && gx == 128) rhs += srcterm;
        next[idx] = inv_den * rhs;
    }
}

extern "C" void kernel_launch(void* const* d_in, const int* in_sizes, int n_in,
                              void* d_out, int out_size, void* d_ws, size_t ws_size,
                              hipStream_t stream) {
    const float* u1    = (const float*)d_in[0];
    const float* u0    = (const float*)d_in[1];
    const float* epsr  = (const float*)d_in[2];
    const float* sigma = (const float*)d_in[3];
    const float* fdk   = (const float*)d_in[4];
    const float* cw    = (const float*)d_in[5];
    const float* src   = (const float*)d_in[6];
    float* out = (float*)d_out;
    float* ws  = (float*)d_ws;

    float* bufs[3] = {ws, ws + NELEM, ws + 2 * NELEM};
    float* Sbuf = ws + 3 * NELEM;  // 2688 floats

    fdtd_init<<<1, 256, 0, stream>>>(fdk, cw, Sbuf);

    const float* cur  = u1;  // carry = (u0, u1): cur=u1, prev=u0
    const float* prev = u0;
    dim3 grid(NBATCH, 16, 2);
    for (int s = 0; s < 32; ++s) {
        float* next = (s == 31) ? out : bufs[s % 3];  // 3-way rotation: no aliasing
        fdtd_step<<<grid, 256, 0, stream>>>(cur, prev, epsr, sigma, Sbuf, src, next, s);
        prev = cur;
        cur  = next;
    }
}